// AttentionDecoder_72215580115714
// MI455X (gfx1250) — compile-verified
//
#include <hip/hip_runtime.h>
#include <math.h>

// ---------------------------------------------------------------------------
// MI455X (gfx1250) attention-decoder step.
// Batch-1 -> all GEMVs, ~800MB weight reads per call => HBM-bound (~34us at
// 23.3 TB/s). GEMVs are mapped onto V_WMMA_F32_16X16X4_F32: A = 16x4 weight
// tile (per-lane contiguous row streaming), B = x broadcast across N columns,
// so D's columns all hold the 16 dot products. K is split across 4 waves per
// 16-row block, combined through LDS. No atomics -> deterministic.
// Inner loop: pointer-increment addressing (immediate-offset loads), 8 loads
// + 8 WMMAs per trip, plus global_prefetch_b8 1KB ahead for deeper MLP.
// ---------------------------------------------------------------------------

typedef __attribute__((ext_vector_type(2))) float v2f;
typedef __attribute__((ext_vector_type(4))) float v4f;
typedef __attribute__((ext_vector_type(8))) float v8f;

#define HID 4096
#define LSEQ 2048

// ---- WMMA GEMV core --------------------------------------------------------
// Accumulates y[row0..row0+15] partials for K-range [kbeg,kend) of  W(N,K)*x.
// A layout (32-bit 16x4): lane m<16 holds (K=k, K=k+1) of row m; lane m+16
// holds (K=k+2, K=k+3) of row m.  B assumed mirrored: lane n holds column n,
// VGPR0 = {K=k | K=k+2}, VGPR1 = {K=k+1 | K=k+3} by lane half.
__device__ __forceinline__ void gemv_slice(const float* __restrict__ W,
                                           const float* __restrict__ x,
                                           int K, int row, int koff, bool lo,
                                           int kbeg, int kend,
                                           v8f& acc0, v8f& acc1)
{
  const float* wp = W + (size_t)row * (size_t)K + koff + kbeg;
  const float* xp = x + kbeg;
  int iters = (kend - kbeg) >> 5;            // 32 K-values per trip
  for (int i = 0; i < iters; ++i) {
    // 8 A-tiles (immediate-offset b64 loads from this lane's row stream)
    v2f a0 = *(const v2f*)(wp + 0);
    v2f a1 = *(const v2f*)(wp + 4);
    v2f a2 = *(const v2f*)(wp + 8);
    v2f a3 = *(const v2f*)(wp + 12);
    v2f a4 = *(const v2f*)(wp + 16);
    v2f a5 = *(const v2f*)(wp + 20);
    v2f a6 = *(const v2f*)(wp + 24);
    v2f a7 = *(const v2f*)(wp + 28);
    // x broadcast (uniform address -> cache/SMEM friendly)
    v4f q0 = *(const v4f*)(xp + 0);
    v4f q1 = *(const v4f*)(xp + 4);
    v4f q2 = *(const v4f*)(xp + 8);
    v4f q3 = *(const v4f*)(xp + 12);
    v4f q4 = *(const v4f*)(xp + 16);
    v4f q5 = *(const v4f*)(xp + 20);
    v4f q6 = *(const v4f*)(xp + 24);
    v4f q7 = *(const v4f*)(xp + 28);
    // stream-ahead prefetch (~1KB) -> global_prefetch_b8
    __builtin_prefetch(wp + 256, 0, 1);

    v2f b0 = lo ? (v2f){q0.x, q0.y} : (v2f){q0.z, q0.w};
    v2f b1 = lo ? (v2f){q1.x, q1.y} : (v2f){q1.z, q1.w};
    v2f b2 = lo ? (v2f){q2.x, q2.y} : (v2f){q2.z, q2.w};
    v2f b3 = lo ? (v2f){q3.x, q3.y} : (v2f){q3.z, q3.w};
    v2f b4 = lo ? (v2f){q4.x, q4.y} : (v2f){q4.z, q4.w};
    v2f b5 = lo ? (v2f){q5.x, q5.y} : (v2f){q5.z, q5.w};
    v2f b6 = lo ? (v2f){q6.x, q6.y} : (v2f){q6.z, q6.w};
    v2f b7 = lo ? (v2f){q7.x, q7.y} : (v2f){q7.z, q7.w};

    // 8 args: (neg_a, A, neg_b, B, c_mod, C, reuse_a, reuse_b)
    acc0 = __builtin_amdgcn_wmma_f32_16x16x4_f32(false, a0, false, b0,
                                                 (short)0, acc0, false, false);
    acc1 = __builtin_amdgcn_wmma_f32_16x16x4_f32(false, a1, false, b1,
                                                 (short)0, acc1, false, false);
    acc0 = __builtin_amdgcn_wmma_f32_16x16x4_f32(false, a2, false, b2,
                                                 (short)0, acc0, false, false);
    acc1 = __builtin_amdgcn_wmma_f32_16x16x4_f32(false, a3, false, b3,
                                                 (short)0, acc1, false, false);
    acc0 = __builtin_amdgcn_wmma_f32_16x16x4_f32(false, a4, false, b4,
                                                 (short)0, acc0, false, false);
    acc1 = __builtin_amdgcn_wmma_f32_16x16x4_f32(false, a5, false, b5,
                                                 (short)0, acc1, false, false);
    acc0 = __builtin_amdgcn_wmma_f32_16x16x4_f32(false, a6, false, b6,
                                                 (short)0, acc0, false, false);
    acc1 = __builtin_amdgcn_wmma_f32_16x16x4_f32(false, a7, false, b7,
                                                 (short)0, acc1, false, false);
    wp += 32;
    xp += 32;
  }
}

// One block = 16 output rows, 4 waves splitting K. Optional second GEMV
// (for gates = W_ih*x + W_hh*h) accumulated in the same pass.
// ACT: 0 = identity, 1 = relu.
template <int ACT>
__global__ __launch_bounds__(128) void gemv_wmma_kernel(
    const float* __restrict__ W1, const float* __restrict__ x1, int K1,
    const float* __restrict__ W2, const float* __restrict__ x2, int K2,
    const float* __restrict__ b1, const float* __restrict__ b2,
    float* __restrict__ out)
{
  __shared__ float red[4 * 16];
  const int tid  = threadIdx.x;
  const int wv   = tid >> 5;        // wave 0..3 (wave32)
  const int lane = tid & 31;
  const int r0   = blockIdx.x * 16;
  const int row  = r0 + (lane & 15);
  const int koff = (lane >> 4) << 1;   // 0 or 2
  const bool lo  = (koff == 0);

  v8f acc0 = {0.f,0.f,0.f,0.f,0.f,0.f,0.f,0.f};
  v8f acc1 = {0.f,0.f,0.f,0.f,0.f,0.f,0.f,0.f};

  {
    int kc = K1 >> 2;                 // K slice per wave (multiple of 32)
    gemv_slice(W1, x1, K1, row, koff, lo, wv * kc, wv * kc + kc, acc0, acc1);
  }
  if (W2) {
    int kc = K2 >> 2;
    gemv_slice(W2, x2, K2, row, koff, lo, wv * kc, wv * kc + kc, acc0, acc1);
  }
  v8f acc = acc0 + acc1;

  // C/D layout: VGPR v, lanes 0-15 -> M=v; lanes 16-31 -> M=8+v. Column N is
  // the lane index and every column is identical, so lane 0 / lane 16 carry
  // rows r0..r0+7 / r0+8..r0+15.
  if ((lane & 15) == 0) {
    int half = lane >> 4;
#pragma unroll
    for (int v = 0; v < 8; ++v) red[wv * 16 + half * 8 + v] = acc[v];
  }
  __syncthreads();
  if (tid < 16) {
    float s = red[tid] + red[16 + tid] + red[32 + tid] + red[48 + tid];
    s += b1[r0 + tid];
    if (b2) s += b2[r0 + tid];
    if (ACT == 1) s = fmaxf(s, 0.f);
    out[r0 + tid] = s;
  }
}

// ---- pre/post kernels ------------------------------------------------------

// Build eh = [e | h] and ec[0:H] = e  (embedding row gather).
__global__ void setup_kernel(const int* __restrict__ token,
                             const float* __restrict__ emb,
                             const float* __restrict__ h,
                             float* __restrict__ eh, float* __restrict__ ec)
{
  int j = blockIdx.x * blockDim.x + threadIdx.x;
  float e = emb[(size_t)token[0] * HID + j];
  eh[j] = e;
  ec[j] = e;
  eh[HID + j] = h[j];
}

// softmax over 2048 logits -> aw (ws) and d_out attn slot.
__global__ void softmax2048_kernel(const float* __restrict__ z,
                                   float* __restrict__ aw,
                                   float* __restrict__ out_aw)
{
  __shared__ float sm[1024];
  int t = threadIdx.x;
  float z0 = z[t], z1 = z[t + 1024];
  sm[t] = fmaxf(z0, z1);
  __syncthreads();
  for (int s = 512; s > 0; s >>= 1) {
    if (t < s) sm[t] = fmaxf(sm[t], sm[t + s]);
    __syncthreads();
  }
  float mx = sm[0];
  __syncthreads();
  float e0 = expf(z0 - mx), e1 = expf(z1 - mx);
  sm[t] = e0 + e1;
  __syncthreads();
  for (int s = 512; s > 0; s >>= 1) {
    if (t < s) sm[t] += sm[t + s];
    __syncthreads();
  }
  float inv = 1.f / sm[0];
  float a0 = e0 * inv, a1 = e1 * inv;
  aw[t] = a0;          aw[t + 1024] = a1;
  out_aw[t] = a0;      out_aw[t + 1024] = a1;
}

// ctx partials: block (jb,ls) -> part[ls][j] = sum_{l in slice} aw[l]*E[l][j]
// Coalesced: 256 consecutive j per block; l uniform -> scalar aw load.
__global__ void ctx_part_kernel(const float* __restrict__ aw,
                                const float* __restrict__ enc,
                                float* __restrict__ part)
{
  int jb = blockIdx.x & 15;
  int ls = blockIdx.x >> 4;           // 0..7
  int j  = jb * 256 + threadIdx.x;
  int l0 = ls * 256;
  float acc = 0.f;
#pragma unroll 4
  for (int l = l0; l < l0 + 256; ++l)
    acc = fmaf(aw[l], enc[(size_t)l * HID + j], acc);
  part[ls * HID + j] = acc;
}

// ec[H + j] = sum of 8 partials (fixed order -> deterministic).
__global__ void ctx_reduce_kernel(const float* __restrict__ part,
                                  float* __restrict__ ec_ctx)
{
  int j = blockIdx.x * blockDim.x + threadIdx.x;
  float s = 0.f;
#pragma unroll
  for (int w = 0; w < 8; ++w) s += part[w * HID + j];
  ec_ctx[j] = s;
}

__device__ __forceinline__ float sigmoidf(float v) {
  return 1.f / (1.f + expf(-v));
}

// LSTM cell elementwise (PyTorch gate order i,f,g,o).
__global__ void lstm_kernel(const float* __restrict__ gates,
                            const float* __restrict__ c_in,
                            float* __restrict__ d_out,   // [logp|h|c|attn]
                            float* __restrict__ hn)
{
  int j = blockIdx.x * blockDim.x + threadIdx.x;
  float ig = gates[j];
  float fg = gates[HID + j];
  float gg = gates[2 * HID + j];
  float og = gates[3 * HID + j];
  float cn = sigmoidf(fg) * c_in[j] + sigmoidf(ig) * tanhf(gg);
  float hv = sigmoidf(og) * tanhf(cn);
  d_out[2 * HID + j] = cn;   // c_new
  d_out[HID + j]     = hv;   // h_new
  hn[j] = hv;
}

// log_softmax over 4096 logits -> d_out[0:4096].
__global__ void logsoftmax4096_kernel(const float* __restrict__ z,
                                      float* __restrict__ out)
{
  __shared__ float sm[1024];
  int t = threadIdx.x;
  float v0 = z[t], v1 = z[t + 1024], v2 = z[t + 2048], v3 = z[t + 3072];
  sm[t] = fmaxf(fmaxf(v0, v1), fmaxf(v2, v3));
  __syncthreads();
  for (int s = 512; s > 0; s >>= 1) {
    if (t < s) sm[t] = fmaxf(sm[t], sm[t + s]);
    __syncthreads();
  }
  float mx = sm[0];
  __syncthreads();
  sm[t] = expf(v0 - mx) + expf(v1 - mx) + expf(v2 - mx) + expf(v3 - mx);
  __syncthreads();
  for (int s = 512; s > 0; s >>= 1) {
    if (t < s) sm[t] += sm[t + s];
    __syncthreads();
  }
  float lse = mx + logf(sm[0]);
  out[t]        = v0 - lse;
  out[t + 1024] = v1 - lse;
  out[t + 2048] = v2 - lse;
  out[t + 3072] = v3 - lse;
}

// ---------------------------------------------------------------------------
extern "C" void kernel_launch(void* const* d_in, const int* in_sizes, int n_in,
                              void* d_out, int out_size, void* d_ws, size_t ws_size,
                              hipStream_t stream) {
  const int*   token  = (const int*)  d_in[0];
  const float* h      = (const float*)d_in[1];
  const float* c      = (const float*)d_in[2];
  const float* enc    = (const float*)d_in[3];
  const float* emb    = (const float*)d_in[4];
  const float* W_attn = (const float*)d_in[5];
  const float* b_attn = (const float*)d_in[6];
  const float* W_comb = (const float*)d_in[7];
  const float* b_comb = (const float*)d_in[8];
  const float* W_ih   = (const float*)d_in[9];
  const float* W_hh   = (const float*)d_in[10];
  const float* b_ih   = (const float*)d_in[11];
  const float* b_hh   = (const float*)d_in[12];
  const float* W_out  = (const float*)d_in[13];
  const float* b_out  = (const float*)d_in[14];
  float* out = (float*)d_out;   // [logp 4096 | h_new 4096 | c_new 4096 | attn 2048]

  float* ws   = (float*)d_ws;
  float* eh     = ws;                //  8192: [e | h]
  float* ec     = ws + 8192;         //  8192: [e | ctx]
  float* zbuf   = ws + 16384;        //  2048: attn logits
  float* aw     = ws + 18432;        //  2048: attn weights
  float* xbuf   = ws + 20480;        //  4096: LSTM input x
  float* gates  = ws + 24576;        // 16384
  float* hn     = ws + 40960;        //  4096: h_new
  float* logits = ws + 45056;        //  4096
  float* part   = ws + 49152;        //  8*4096 ctx partials

  // 1) embedding gather + concat staging
  setup_kernel<<<16, 256, 0, stream>>>(token, emb, h, eh, ec);

  // 2) attn logits: [e,h](8192) x W_attn(2048x8192)  -> z
  gemv_wmma_kernel<0><<<LSEQ / 16, 128, 0, stream>>>(
      W_attn, eh, 2 * HID, nullptr, nullptr, 0, b_attn, nullptr, zbuf);

  // 3) softmax(2048)
  softmax2048_kernel<<<1, 1024, 0, stream>>>(zbuf, aw, out + 3 * HID);

  // 4) ctx = aw @ encoder_outputs  (partials over l, then reduce)
  ctx_part_kernel<<<16 * 8, 256, 0, stream>>>(aw, enc, part);
  ctx_reduce_kernel<<<16, 256, 0, stream>>>(part, ec + HID);

  // 5) x = relu([e,ctx] x W_comb(4096x8192) + b_comb)
  gemv_wmma_kernel<1><<<HID / 16, 128, 0, stream>>>(
      W_comb, ec, 2 * HID, nullptr, nullptr, 0, b_comb, nullptr, xbuf);

  // 6) gates = x*W_ih^T + h*W_hh^T + b_ih + b_hh   (dual GEMV, 16384 rows)
  gemv_wmma_kernel<0><<<(4 * HID) / 16, 128, 0, stream>>>(
      W_ih, xbuf, HID, W_hh, eh + HID, HID, b_ih, b_hh, gates);

  // 7) LSTM cell -> c_new, h_new
  lstm_kernel<<<16, 256, 0, stream>>>(gates, c, out, hn);

  // 8) logits = h_new x W_out(4096x4096) + b_out
  gemv_wmma_kernel<0><<<HID / 16, 128, 0, stream>>>(
      W_out, hn, HID, nullptr, nullptr, 0, b_out, nullptr, logits);

  // 9) log_softmax -> logp
  logsoftmax4096_kernel<<<1, 1024, 0, stream>>>(logits, out);
}